// ScaledDotProductAttention_84164179133075
// MI455X (gfx1250) — compile-verified
//
#include <hip/hip_runtime.h>
#include <hip/hip_bf16.h>

// Problem constants (from reference)
#define B_      64
#define L_      4096
#define DK_     128
#define DV_     128
#define H_      4
#define DKL_    64
#define R_      128

// Tiling
#define NCHUNK    16                 // L split into 16 chunks of 256 rows
#define CH_ROWS   (L_ / NCHUNK)      // 256 rows per workgroup
#define TILES     (CH_ROWS / 16)     // 16 row-tiles of 16
#define PSTR      (2*H_ + H_*DV_)    // 520 floats per (b,chunk) partial: M[4],Z[4],A[4][128]
#define SCALE_    0.08838834764831845f  // 1/sqrt(128)

typedef float v2f __attribute__((ext_vector_type(2)));
typedef float v8f __attribute__((ext_vector_type(8)));

// ---------------------------------------------------------------------------
// Kernel A: q[h][d] = sum_k Wq[h][k] * Wk[d][h][k]   (tiny, 1 block)
// ---------------------------------------------------------------------------
__global__ void sdpa_qcomb(const float* __restrict__ Wq,
                           const float* __restrict__ Wk,
                           float* __restrict__ qv) {
  const int t = threadIdx.x;
  if (t < H_ * DK_) {
    const int h = t >> 7, d = t & 127;
    float sum = 0.0f;
    #pragma unroll 8
    for (int k = 0; k < DKL_; ++k)
      sum += Wq[(h << 6) + k] * Wk[(size_t)(d * H_ + h) * DKL_ + k];
    qv[(h << 7) + d] = sum;
  }
}

// ---------------------------------------------------------------------------
// Kernel B: streaming flash pass over K and V (the 268 MB of traffic).
// Per 16-row tile: logits = K_tile(16x128) @ q^T(128x4, padded to 16 cols)
// via 32 chained V_WMMA_F32_16X16X4_F32, then online softmax + V accumulate.
// ---------------------------------------------------------------------------
__global__ __launch_bounds__(256) void sdpa_flash(
    const float* __restrict__ K, const float* __restrict__ V,
    const unsigned char* __restrict__ mask,  // numpy bool_: 1 byte per element
    const float* __restrict__ qv, float* __restrict__ partial) {
  __shared__ float qs[16][128];      // q^T padded to 16 "heads" (rows 4..15 = 0)
  __shared__ float tl[8][16][4];     // per-wave tile logits broadcast
  __shared__ float redA[8][4][128];  // cross-wave reduction
  __shared__ float redMS[8][2][4];
  __shared__ float Mc[4];

  const int bc = blockIdx.x;
  const int b = bc / NCHUNK, chunk = bc % NCHUNK;
  const int t = threadIdx.x;
  const int wave = t >> 5, lane = t & 31;
  const int hi = lane >> 4, lo = lane & 15;

  for (int i = t; i < 16 * 128; i += 256) {
    const int row = i >> 7, col = i & 127;
    qs[row][col] = (row < H_) ? qv[(row << 7) + col] : 0.0f;
  }
  __syncthreads();

  float m[H_], s[H_];
  float4 acc[H_];
  #pragma unroll
  for (int h = 0; h < H_; ++h) {
    m[h] = -1.0e30f; s[h] = 0.0f;
    acc[h] = make_float4(0.f, 0.f, 0.f, 0.f);
  }

  const size_t baseK = (size_t)b * L_ * DK_;
  const size_t baseV = (size_t)b * L_ * DV_;
  const int l0c = chunk * CH_ROWS;

  for (int tile = wave; tile < TILES; tile += 8) {   // 2 tiles per wave
    const int l0 = l0c + (tile << 4);

    // --- logits via chained f32 WMMA over K-dim (DK=128 => 32 steps of k=4)
    // A layout (16x4 f32): lane m=lo holds row m; VGPR pair covers k = 2*hi+{0,1}
    // B layout (4x16 f32): lane n=lo holds col n; same k sub-indexing
    const float2* a2 = (const float2*)(K + baseK + (size_t)(l0 + lo) * DK_);
    const float2* b2 = (const float2*)(&qs[lo][0]);
    v8f c = {0.f, 0.f, 0.f, 0.f, 0.f, 0.f, 0.f, 0.f};
    #pragma unroll
    for (int kb = 0; kb < 32; ++kb) {
      const float2 af = a2[2 * kb + hi];
      const float2 bf = b2[2 * kb + hi];
      v2f a;  a[0] = af.x;  a[1] = af.y;
      v2f bb; bb[0] = bf.x; bb[1] = bf.y;
      c = __builtin_amdgcn_wmma_f32_16x16x4_f32(false, a, false, bb,
                                                (short)0, c, false, false);
    }

    // --- scale + mask, broadcast tile logits through LDS
    // C/D layout: VGPR j -> row j (lanes 0-15) / row j+8 (lanes 16-31), col = lo
    if (lo < H_) {
      #pragma unroll
      for (int j = 0; j < 8; ++j) {
        const int mrow = (hi << 3) + j;
        const float lg = mask[(size_t)b * L_ + l0 + mrow] ? -INFINITY
                                                          : c[j] * SCALE_;
        tl[wave][mrow][lo] = lg;
      }
    }
    asm volatile("s_wait_dscnt 0" ::: "memory");  // intra-wave LDS visibility

    // --- online softmax update (finite floor avoids (-inf)-(-inf) NaN)
    float tmax[H_];
    #pragma unroll
    for (int h = 0; h < H_; ++h) tmax[h] = -1.0e30f;
    for (int j = 0; j < 16; ++j) {
      #pragma unroll
      for (int h = 0; h < H_; ++h) tmax[h] = fmaxf(tmax[h], tl[wave][j][h]);
    }
    #pragma unroll
    for (int h = 0; h < H_; ++h) {
      const float mn = fmaxf(m[h], tmax[h]);
      const float corr = __expf(m[h] - mn);
      s[h] *= corr;
      acc[h].x *= corr; acc[h].y *= corr; acc[h].z *= corr; acc[h].w *= corr;
      m[h] = mn;
    }

    // --- weighted V accumulation (coalesced float4 per row; lane owns 4 cols)
    for (int j = 0; j < 16; ++j) {
      const float4 v4 =
          ((const float4*)(V + baseV + (size_t)(l0 + j) * DV_))[lane];
      #pragma unroll
      for (int h = 0; h < H_; ++h) {
        const float p = __expf(tl[wave][j][h] - m[h]);
        s[h] += p;
        acc[h].x += p * v4.x; acc[h].y += p * v4.y;
        acc[h].z += p * v4.z; acc[h].w += p * v4.w;
      }
    }
  }

  // --- combine the 8 waves' partials, emit per-(b,chunk) partial
  #pragma unroll
  for (int h = 0; h < H_; ++h) {
    redA[wave][h][4 * lane + 0] = acc[h].x;
    redA[wave][h][4 * lane + 1] = acc[h].y;
    redA[wave][h][4 * lane + 2] = acc[h].z;
    redA[wave][h][4 * lane + 3] = acc[h].w;
  }
  if (lane == 0) {
    #pragma unroll
    for (int h = 0; h < H_; ++h) {
      redMS[wave][0][h] = m[h];
      redMS[wave][1][h] = s[h];
    }
  }
  __syncthreads();

  float* pout = partial + (size_t)bc * PSTR;
  if (t < H_) {
    const int h = t;
    float M = -1.0e30f;
    for (int w = 0; w < 8; ++w) M = fmaxf(M, redMS[w][0][h]);
    float Z = 0.0f;
    for (int w = 0; w < 8; ++w) Z += redMS[w][1][h] * __expf(redMS[w][0][h] - M);
    pout[h] = M;
    pout[H_ + h] = Z;
    Mc[h] = M;
  }
  __syncthreads();
  for (int idx = t; idx < H_ * DV_; idx += 256) {
    const int h = idx >> 7;
    float sum = 0.0f;
    for (int w = 0; w < 8; ++w)
      sum += redA[w][h][idx & 127] * __expf(redMS[w][0][h] - Mc[h]);
    pout[2 * H_ + idx] = sum;
  }
}

// ---------------------------------------------------------------------------
// Kernel C: combine NCHUNK partials per batch, apply Wv (/Z/L) then Wo.
// ---------------------------------------------------------------------------
__global__ __launch_bounds__(256) void sdpa_finalize(
    const float* __restrict__ partial, const float* __restrict__ Wv,
    const float* __restrict__ Wo, float* __restrict__ out) {
  __shared__ float Msh[H_], Zsh[H_];
  __shared__ float Ssh[H_][DV_];
  __shared__ float Qsh[H_][DV_];

  const int b = blockIdx.x, t = threadIdx.x;
  const float* pb = partial + (size_t)b * NCHUNK * PSTR;

  if (t < H_) {
    const int h = t;
    float M = -1.0e30f;
    for (int c = 0; c < NCHUNK; ++c) M = fmaxf(M, pb[c * PSTR + h]);
    float Z = 0.0f;
    for (int c = 0; c < NCHUNK; ++c)
      Z += pb[c * PSTR + H_ + h] * __expf(pb[c * PSTR + h] - M);
    Msh[h] = M; Zsh[h] = Z;
  }
  __syncthreads();

  for (int idx = t; idx < H_ * DV_; idx += 256) {
    const int h = idx >> 7;
    float sum = 0.0f;
    for (int c = 0; c < NCHUNK; ++c)
      sum += pb[c * PSTR + 2 * H_ + idx] * __expf(pb[c * PSTR + h] - Msh[h]);
    Ssh[h][idx & 127] = sum;
  }
  __syncthreads();

  // QKV[h][v] = (1/(L*Z[h])) * sum_e S[h][e] * Wv[e][h][v]
  for (int idx = t; idx < H_ * DV_; idx += 256) {
    const int h = idx >> 7, v = idx & 127;
    float sum = 0.0f;
    #pragma unroll 8
    for (int e = 0; e < DV_; ++e)
      sum += Ssh[h][e] * Wv[(size_t)(e * H_ + h) * DV_ + v];
    Qsh[h][v] = sum / (Zsh[h] * (float)L_);
  }
  __syncthreads();

  // out[b][r] = sum_{h,v} QKV[h][v] * Wo[h][v][r]
  if (t < R_) {
    float sum = 0.0f;
    #pragma unroll
    for (int h = 0; h < H_; ++h)
      for (int v = 0; v < DV_; ++v)
        sum += Qsh[h][v] * Wo[(size_t)(h * DV_ + v) * R_ + t];
    out[(size_t)b * R_ + t] = sum;
  }
}

// ---------------------------------------------------------------------------
extern "C" void kernel_launch(void* const* d_in, const int* in_sizes, int n_in,
                              void* d_out, int out_size, void* d_ws,
                              size_t ws_size, hipStream_t stream) {
  const float* K  = (const float*)d_in[0];
  const float* V  = (const float*)d_in[1];
  const unsigned char* mask = (const unsigned char*)d_in[2];  // jnp bool -> 1B
  const float* Wq = (const float*)d_in[3];
  const float* Wk = (const float*)d_in[4];
  const float* Wv = (const float*)d_in[5];
  const float* Wo = (const float*)d_in[6];
  float* out = (float*)d_out;

  float* qv      = (float*)d_ws;            // 512 floats
  float* partial = qv + H_ * DK_;           // B*NCHUNK*520 floats (~2.1 MB)

  sdpa_qcomb<<<1, 512, 0, stream>>>(Wq, Wk, qv);
  sdpa_flash<<<B_ * NCHUNK, 256, 0, stream>>>(K, V, mask, qv, partial);
  sdpa_finalize<<<B_, 256, 0, stream>>>(partial, Wv, Wo, out);
}